// GNN_56616258896202
// MI455X (gfx1250) — compile-verified
//
#include <hip/hip_runtime.h>
#include <hip/hip_bf16.h>
#include <climits>

typedef float v2f __attribute__((ext_vector_type(2)));
typedef float v8f __attribute__((ext_vector_type(8)));

#define NEG_SLOPE_ATT 0.2f
#define NEG_SLOPE_ACT 0.01f

__device__ __forceinline__ float lrelu(float v, float slope) {
    return v > 0.0f ? v : slope * v;
}

// Monotone float->int mapping so integer atomicMax == float max.
__device__ __forceinline__ int fmap(float f) {
    int i = __float_as_int(f);
    return i >= 0 ? i : (i ^ 0x7fffffff);
}
__device__ __forceinline__ float funmap(int i) {
    return __int_as_float(i >= 0 ? i : (i ^ 0x7fffffff));
}

// --------------------------------------------------------------------------
// C[M x NC] = A[M x KC] * B[KC x NC], row-major fp32, KC/NC compile-time.
// B is staged in LDS once per block (W1 = 64KB, W2 = 32KB; WGP LDS = 320KB).
// One wave32 computes a 16x16 tile via V_WMMA_F32_16X16X4_F32 (K step = 4):
//   A frag (16x4): lanes 0-15 row M=lane K={k0,k0+1}; lanes 16-31 K={k0+2,k0+3}
//   B frag (4x16): vgpr j: lanes 0-15 K=k0+j; lanes 16-31 K=k0+2+j
//   C/D  (16x16): vgpr v: lanes 0-15 M=v; lanes 16-31 M=v+8
// --------------------------------------------------------------------------
template <int KC, int NC>
__global__ void gat_gemm_wmma_f32(const float* __restrict__ A,
                                  const float* __restrict__ B,
                                  float* __restrict__ C, int M) {
    __shared__ float Bs[KC * NC];

    const int tid = threadIdx.x;
    // Cooperative, coalesced load of the whole weight matrix into LDS.
#pragma unroll
    for (int i = tid; i < KC * NC; i += 256) Bs[i] = B[i];
    __syncthreads();

    const int lane = tid & 31;
    const int wave = tid >> 5;
    const int half = lane >> 4;   // 0: lanes 0-15, 1: lanes 16-31
    const int l15  = lane & 15;
    const int mbase = blockIdx.x << 4;
    if (mbase >= M) return;

    int arow = mbase + l15;
    if (arow >= M) arow = M - 1;                    // clamp (keeps EXEC full)
    const float* aptr = A + (size_t)arow * KC + 2 * half;

    constexpr int NTILES = NC >> 4;                 // 8 (layer1) or 4 (layer2)
    for (int nt = wave; nt < NTILES; nt += 8) {
        const float* bs = Bs + (nt << 4) + l15 + 2 * half * NC;
        v8f acc = {};
#pragma unroll
        for (int k0 = 0; k0 < KC; k0 += 4) {
            v2f a;
            a.x = aptr[k0];
            a.y = aptr[k0 + 1];
            v2f b;
            b.x = bs[k0 * NC];          // immediate DS offsets after unroll
            b.y = bs[k0 * NC + NC];
            acc = __builtin_amdgcn_wmma_f32_16x16x4_f32(
                false, a, false, b, (short)0, acc, false, false);
        }
#pragma unroll
        for (int v = 0; v < 8; ++v) {
            const int row = mbase + v + 8 * half;
            if (row < M) C[(size_t)row * NC + (nt << 4) + l15] = acc[v];
        }
    }
}

// Per-node attention logits: s[n] = h[n,:]·a_src, d[n] = h[n,:]·a_dst. Wave/node.
template <int F>
__global__ void gat_logits(const float* __restrict__ h,
                           const float* __restrict__ a_src,
                           const float* __restrict__ a_dst,
                           float* __restrict__ s, float* __restrict__ d, int N) {
    const int lane = threadIdx.x & 31;
    const int node = blockIdx.x * (blockDim.x >> 5) + (threadIdx.x >> 5);
    if (node >= N) return;
    const float* row = h + (size_t)node * F;
    float ss = 0.0f, dd = 0.0f;
#pragma unroll
    for (int f = lane; f < F; f += 32) {
        float v = row[f];
        ss += v * a_src[f];
        dd += v * a_dst[f];
    }
#pragma unroll
    for (int off = 16; off > 0; off >>= 1) {
        ss += __shfl_down(ss, off, 32);
        dd += __shfl_down(dd, off, 32);
    }
    if (lane == 0) { s[node] = ss; d[node] = dd; }
}

// Zero accumulator [N*F], init segment-max to INT_MIN, denom to 0.
__global__ void gat_init(float* __restrict__ acc, int* __restrict__ m,
                         float* __restrict__ denom, int N, int F) {
    const int total = N * F;
    for (int i = blockIdx.x * blockDim.x + threadIdx.x; i < total;
         i += gridDim.x * blockDim.x) {
        acc[i] = 0.0f;
        if (i < N) { m[i] = INT_MIN; denom[i] = 0.0f; }
    }
}

// Pass 1: segment max of edge logits over dst (self-loops appended virtually).
__global__ void gat_edge_max(const int* __restrict__ ei,
                             const float* __restrict__ s,
                             const float* __restrict__ d,
                             int* __restrict__ m, int E, int N) {
    const int total = E + N;
    for (int e = blockIdx.x * blockDim.x + threadIdx.x; e < total;
         e += gridDim.x * blockDim.x) {
        const int src = (e < E) ? ei[e]     : (e - E);
        const int dst = (e < E) ? ei[E + e] : (e - E);
        const float v = lrelu(s[src] + d[dst], NEG_SLOPE_ATT);
        atomicMax(&m[dst], fmap(v));
    }
}

// Pass 2: ex = exp(e - max); denom[dst] += ex; acc[dst,:] += h[src,:] * ex.
// One wave per edge; each lane owns F/32 contiguous features (vectorizable).
template <int F>
__global__ void gat_edge_scatter(const int* __restrict__ ei,
                                 const float* __restrict__ s,
                                 const float* __restrict__ d,
                                 const int* __restrict__ m,
                                 const float* __restrict__ h,
                                 float* __restrict__ denom,
                                 float* __restrict__ acc, int E, int N) {
    constexpr int PER = F / 32;   // 4 (layer1) or 2 (layer2)
    const int lane = threadIdx.x & 31;
    const int wavesPerBlock = blockDim.x >> 5;
    const int nwaves = gridDim.x * wavesPerBlock;
    const int total = E + N;
    for (int e = blockIdx.x * wavesPerBlock + (threadIdx.x >> 5); e < total;
         e += nwaves) {
        const int src = (e < E) ? ei[e]     : (e - E);
        const int dst = (e < E) ? ei[E + e] : (e - E);
        const float v  = lrelu(s[src] + d[dst], NEG_SLOPE_ATT);
        const float ex = __expf(v - funmap(m[dst]));
        if (lane == 0) atomicAdd(&denom[dst], ex);
        const float* hrow = h   + (size_t)src * F + lane * PER;
        float*       arow = acc + (size_t)dst * F + lane * PER;
        float hv[PER];
#pragma unroll
        for (int j = 0; j < PER; ++j) hv[j] = hrow[j];   // merges to b128/b64
#pragma unroll
        for (int j = 0; j < PER; ++j) atomicAdd(&arow[j], hv[j] * ex);
    }
}

// out = leaky_relu(acc/denom + bias, 0.01)  (layer-1 finalize -> layer-2 input)
template <int F>
__global__ void gat_finalize_act(const float* __restrict__ acc,
                                 const float* __restrict__ denom,
                                 const float* __restrict__ bias,
                                 float* __restrict__ out, int N) {
    const int total = N * F;
    for (int i = blockIdx.x * blockDim.x + threadIdx.x; i < total;
         i += gridDim.x * blockDim.x) {
        const int n = i / F, f = i & (F - 1);
        out[i] = lrelu(acc[i] / denom[n] + bias[f], NEG_SLOPE_ACT);
    }
}

__global__ void gat_pool_zero(float* __restrict__ sums, float* __restrict__ counts,
                              int G, int F) {
    const int total = G * F;
    for (int i = blockIdx.x * blockDim.x + threadIdx.x; i < total;
         i += gridDim.x * blockDim.x) {
        sums[i] = 0.0f;
        if (i < G) counts[i] = 0.0f;
    }
}

// Layer-2 finalize fused with mean-pool accumulation.
template <int F>
__global__ void gat_finalize_pool(const float* __restrict__ acc,
                                  const float* __restrict__ denom,
                                  const float* __restrict__ bias,
                                  const int* __restrict__ batch,
                                  float* __restrict__ sums,
                                  float* __restrict__ counts, int N) {
    const int total = N * F;
    for (int i = blockIdx.x * blockDim.x + threadIdx.x; i < total;
         i += gridDim.x * blockDim.x) {
        const int n = i / F, f = i & (F - 1);
        const float v = acc[i] / denom[n] + bias[f];
        const int g = batch[n];
        atomicAdd(&sums[(size_t)g * F + f], v);
        if (f == 0) atomicAdd(&counts[g], 1.0f);
    }
}

__global__ void gat_pool_div(const float* __restrict__ sums,
                             const float* __restrict__ counts,
                             float* __restrict__ out, int G, int F) {
    const int total = G * F;
    for (int i = blockIdx.x * blockDim.x + threadIdx.x; i < total;
         i += gridDim.x * blockDim.x) {
        const int g = i / F;
        out[i] = sums[i] / fmaxf(counts[g], 1.0f);
    }
}

extern "C" void kernel_launch(void* const* d_in, const int* in_sizes, int n_in,
                              void* d_out, int out_size, void* d_ws, size_t ws_size,
                              hipStream_t stream) {
    const int D = 128, H = 128, O = 64;
    const int N = in_sizes[0] / D;
    const int E = in_sizes[1] / 2;
    const int G = out_size / O;

    const float* x      = (const float*)d_in[0];
    const int*   ei     = (const int*)  d_in[1];
    const int*   batch  = (const int*)  d_in[2];
    const float* W1     = (const float*)d_in[4];
    const float* a_src1 = (const float*)d_in[5];
    const float* a_dst1 = (const float*)d_in[6];
    const float* b1     = (const float*)d_in[7];
    const float* W2     = (const float*)d_in[8];
    const float* a_src2 = (const float*)d_in[9];
    const float* a_dst2 = (const float*)d_in[10];
    const float* b2     = (const float*)d_in[11];
    float* out = (float*)d_out;

    // Workspace layout (floats). h1 is reused as layer-2 input; h2/acc2 live
    // inside the (freed) acc1 region.
    float* ws    = (float*)d_ws;
    float* h1    = ws;                      // N*H   (later: activated layer-2 input)
    float* acc1  = ws + (size_t)N * H;      // N*H
    float* h2    = acc1;                    // N*O   (after acc1 is consumed)
    float* acc2  = acc1 + (size_t)N * O;    // N*O
    float* s     = ws + (size_t)2 * N * H;  // N
    float* dl    = s + N;                   // N
    float* denom = dl + N;                  // N
    int*   m     = (int*)(denom + N);       // N
    float* sums  = (float*)(m + N);         // G*O
    float* cnts  = sums + (size_t)G * O;    // G

    const int T = 256;
    const int totE = E + N;
    const int gNF1   = (N * H + T - 1) / T;
    const int gNF2   = (N * O + T - 1) / T;
    const int gGemm  = (N + 15) / 16;
    const int gWave  = (N + 7) / 8;           // wave-per-node, 8 waves/block
    const int gEdge  = (totE + T - 1) / T;
    const int gEdgeW = (totE + 7) / 8;        // wave-per-edge

    // ---------------- Layer 1 ----------------
    gat_gemm_wmma_f32<128, 128><<<gGemm, T, 0, stream>>>(x, W1, h1, N);
    gat_logits<128><<<gWave, T, 0, stream>>>(h1, a_src1, a_dst1, s, dl, N);
    gat_init<<<gNF1, T, 0, stream>>>(acc1, m, denom, N, H);
    gat_edge_max<<<gEdge, T, 0, stream>>>(ei, s, dl, m, E, N);
    gat_edge_scatter<128><<<gEdgeW, T, 0, stream>>>(ei, s, dl, m, h1, denom, acc1, E, N);
    gat_finalize_act<128><<<gNF1, T, 0, stream>>>(acc1, denom, b1, h1, N);

    // ---------------- Layer 2 ----------------
    gat_gemm_wmma_f32<128, 64><<<gGemm, T, 0, stream>>>(h1, W2, h2, N);
    gat_logits<64><<<gWave, T, 0, stream>>>(h2, a_src2, a_dst2, s, dl, N);
    gat_init<<<gNF2, T, 0, stream>>>(acc2, m, denom, N, O);
    gat_pool_zero<<<(G * O + T - 1) / T, T, 0, stream>>>(sums, cnts, G, O);
    gat_edge_max<<<gEdge, T, 0, stream>>>(ei, s, dl, m, E, N);
    gat_edge_scatter<64><<<gEdgeW, T, 0, stream>>>(ei, s, dl, m, h2, denom, acc2, E, N);
    gat_finalize_pool<64><<<gNF2, T, 0, stream>>>(acc2, denom, b2, batch, sums, cnts, N);
    gat_pool_div<<<(G * O + T - 1) / T, T, 0, stream>>>(sums, cnts, out, G, O);
}